// PDNM_17111149707551
// MI455X (gfx1250) — compile-verified
//
#include <hip/hip_runtime.h>
#include <math.h>

// Problem constants (reference: n=4, c=128, h=w=256)
#define NB   4
#define CIN  128
#define CC   64
#define HF   256     // full resolution
#define HD   64      // h/4
#define HQ   32      // h/8
#define QN   4096    // HD*HD
#define VN   1024    // HQ*HQ
#define BN_EPS 1e-5f
#define EPS_D  1e-6f

typedef float v2f __attribute__((ext_vector_type(2)));
typedef float v8f __attribute__((ext_vector_type(8)));

// --------------------------------------------------------------------------
// Generic strided-batched fp32 WMMA GEMM:
//   C[b][m][n] = sum_k A[b][m][k] * B[b][k][n] (+ bias[m])
// Each wave owns a 16 x (16*NT) output strip: one A fragment feeds NT WMMAs
// per K-step (K in steps of 4 via V_WMMA_F32_16X16X4_F32). All addressing is
// pointer-increment form so the hot loop is loads + v_wmma + adds only.
// Register layout (ISA 7.12.2): A 16x4 -> 2 VGPRs, lanes 0-15 hold K={k,k+1},
// lanes 16-31 hold K={k+2,k+3}, M=lane%16. B 4x16 mirrored with N=lane%16.
// C/D 16x16 -> 8 VGPRs, row = r + 8*(lane>=16), col = lane%16.
// Requires M%16==0, N%(16*NT*waves_per_block)==0, K%4==0 (true below).
// --------------------------------------------------------------------------
template <int NT>
__global__ void gemm16_wmma_f32(
    const float* __restrict__ A, long long a_rs, long long a_cs, long long a_bs,
    const float* __restrict__ B, long long b_rs, long long b_cs, long long b_bs,
    const float* __restrict__ bias,
    float* __restrict__ C, long long c_rs, long long c_cs, long long c_bs,
    int M, int N, int K)
{
    const int lane = threadIdx.x & 31;
    const int wave = threadIdx.x >> 5;
    const int n0 = (blockIdx.x * (blockDim.x >> 5) + wave) * (16 * NT);
    const int m0 = blockIdx.y * 16;
    if (n0 >= N) return;                       // wave-uniform: EXEC stays all-1 for WMMA
    A += (long long)blockIdx.z * a_bs;
    B += (long long)blockIdx.z * b_bs;
    C += (long long)blockIdx.z * c_bs;
    const int lm = lane & 15;
    const int hi = lane >> 4;                  // 0: K pair {0,1}; 1: K pair {2,3}

    v8f acc[NT];
    #pragma unroll
    for (int t = 0; t < NT; ++t) acc[t] = (v8f){0.f,0.f,0.f,0.f,0.f,0.f,0.f,0.f};

    // affine address streams (increment by constant per K-step: no muls in loop)
    const float* ap0 = A + (long long)(m0 + lm) * a_rs + (long long)(hi ? 2 : 0) * a_cs;
    const float* ap1 = A + (long long)(m0 + lm) * a_rs + (long long)(hi ? 3 : 1) * a_cs;
    const float* bp0[NT];
    const float* bp1[NT];
    #pragma unroll
    for (int t = 0; t < NT; ++t) {
        bp0[t] = B + (long long)(hi ? 2 : 0) * b_rs + (long long)(n0 + 16*t + lm) * b_cs;
        bp1[t] = B + (long long)(hi ? 3 : 1) * b_rs + (long long)(n0 + 16*t + lm) * b_cs;
    }
    const long long aStep = 4 * a_cs;
    const long long bStep = 4 * b_rs;

    #pragma unroll 4
    for (int k = 0; k < K; k += 4) {
        v2f a;
        a.x = *ap0;
        a.y = *ap1;
        // near-scope prefetch (locality 3 -> WGP scope: fills all cache levels,
        // ISA 10.5; SYS-scope prefetch would stop at GL2)
        __builtin_prefetch(ap0 + 8 * aStep, 0, 3);       // global_prefetch_b8
        __builtin_prefetch(bp0[0] + 8 * bStep, 0, 3);
        ap0 += aStep; ap1 += aStep;
        #pragma unroll
        for (int t = 0; t < NT; ++t) {
            v2f b;
            b.x = *bp0[t];
            b.y = *bp1[t];
            bp0[t] += bStep; bp1[t] += bStep;
            acc[t] = __builtin_amdgcn_wmma_f32_16x16x4_f32(false, a, false, b,
                                                           (short)0, acc[t], false, false);
        }
    }

    #pragma unroll
    for (int t = 0; t < NT; ++t) {
        #pragma unroll
        for (int r = 0; r < 8; ++r) {
            const int m = m0 + r + hi * 8;
            float v = acc[t][r];
            if (bias) v += bias[m];
            C[(long long)m * c_rs + (long long)(n0 + 16*t + lm) * c_cs] = v;
        }
    }
}

// --------------------------------------------------------------------------
// Depthwise 4x4 stride-4 conv: x[n,c,256,256] -> xd[n,c,64,64]
// --------------------------------------------------------------------------
__global__ void down_conv_kernel(const float* __restrict__ x,
                                 const float* __restrict__ dw,
                                 float* __restrict__ xd)
{
    int idx = blockIdx.x * blockDim.x + threadIdx.x;
    const int total = NB * CIN * HD * HD;
    if (idx >= total) return;
    const int j  = idx % HD;
    const int i  = (idx / HD) % HD;
    const int ch = (idx / (HD * HD)) % CIN;
    const int nb = idx / (HD * HD * CIN);
    const float* xp = x + (((long long)nb * CIN + ch) * HF + i * 4) * HF + j * 4;
    const float* w  = dw + ch * 16;
    float s = 0.f;
    #pragma unroll
    for (int a = 0; a < 4; ++a)
        #pragma unroll
        for (int b = 0; b < 4; ++b)
            s += xp[a * HF + b] * w[a * 4 + b];
    xd[idx] = s;
}

// 2x2 max pool: in [NB*CC, H, H] -> out [NB*CC, H/2, H/2]
__global__ void maxpool2_kernel(const float* __restrict__ in,
                                float* __restrict__ out, int H)
{
    int idx = blockIdx.x * blockDim.x + threadIdx.x;
    const int Ho = H / 2;
    const int total = NB * CC * Ho * Ho;
    if (idx >= total) return;
    const int j  = idx % Ho;
    const int i  = (idx / Ho) % Ho;
    const int nc = idx / (Ho * Ho);
    const float* p = in + ((long long)nc * H + 2 * i) * H + 2 * j;
    out[idx] = fmaxf(fmaxf(p[0], p[1]), fmaxf(p[H], p[H + 1]));
}

// avg pool: in [NB*CC, H, H] -> out [NB*CC, ps, ps], k = H/ps
__global__ void avgpool_kernel(const float* __restrict__ in,
                               float* __restrict__ out, int H, int ps)
{
    int idx = blockIdx.x * blockDim.x + threadIdx.x;
    const int total = NB * CC * ps * ps;
    if (idx >= total) return;
    const int j  = idx % ps;
    const int i  = (idx / ps) % ps;
    const int nc = idx / (ps * ps);
    const int k = H / ps;
    const float* p = in + ((long long)nc * H + i * k) * H + j * k;
    float s = 0.f;
    for (int a = 0; a < k; ++a)
        for (int b = 0; b < k; ++b)
            s += p[a * H + b];
    out[idx] = s / (float)(k * k);
}

// pyramid 1x1 conv (16 out ch, 64 in ch, no bias) + BN(mean0,var1) + ReLU
__global__ void pyrconv_kernel(const float* __restrict__ pool,
                               const float* __restrict__ pw,
                               const float* __restrict__ gamma,
                               const float* __restrict__ beta,
                               float* __restrict__ pyr, int psq)
{
    int idx = blockIdx.x * blockDim.x + threadIdx.x;
    const int total = NB * 16 * psq;
    if (idx >= total) return;
    const int p  = idx % psq;
    const int o  = (idx / psq) % 16;
    const int nb = idx / (16 * psq);
    float s = 0.f;
    for (int c = 0; c < CC; ++c)
        s += pw[o * CC + c] * pool[((long long)nb * CC + c) * psq + p];
    const float bn_scale = rsqrtf(1.0f + BN_EPS);
    s = s * bn_scale * gamma[o] + beta[o];
    pyr[idx] = fmaxf(s, 0.f);
}

// concat [x, up(pyr8), up(pyr4), up(pyr2), up(pyr1)] -> cat [NB, 128, H*H]
// upsample uses align_corners=False (matches F.upsample default in reference)
__global__ void upcat_kernel(const float* __restrict__ xin,
                             const float* __restrict__ pyr,
                             float* __restrict__ cat, int H)
{
    int idx = blockIdx.x * blockDim.x + threadIdx.x;
    const int hw = H * H;
    const int total = NB * 2 * CC * hw;
    if (idx >= total) return;
    const int p  = idx % hw;
    const int ch = (idx / hw) % (2 * CC);
    const int nb = idx / (hw * 2 * CC);
    float v;
    if (ch < CC) {
        v = xin[((long long)nb * CC + ch) * hw + p];
    } else {
        const int i   = (ch - CC) >> 4;
        const int c16 = (ch - CC) & 15;
        const int ps  = 8 >> i;
        const int psq = ps * ps;
        const int offs[4] = {0, NB*16*64, NB*16*(64+16), NB*16*(64+16+4)};
        const float* src = pyr + offs[i] + ((long long)nb * 16 + c16) * psq;
        const int y = p / H, x = p % H;
        float sy = fminf(fmaxf((y + 0.5f) * (float)ps / (float)H - 0.5f, 0.f), ps - 1.f);
        float sx = fminf(fmaxf((x + 0.5f) * (float)ps / (float)H - 0.5f, 0.f), ps - 1.f);
        int y0 = (int)floorf(sy), x0 = (int)floorf(sx);
        int y1 = min(y0 + 1, ps - 1), x1 = min(x0 + 1, ps - 1);
        float fy = sy - y0, fx = sx - x0;
        v = (1.f - fy) * ((1.f - fx) * src[y0*ps+x0] + fx * src[y0*ps+x1])
          +        fy  * ((1.f - fx) * src[y1*ps+x0] + fx * src[y1*ps+x1]);
    }
    cat[idx] = v;
}

// depth map 256x256 -> SxS, align_corners=True
__global__ void depth_resize_kernel(const float* __restrict__ depth,
                                    float* __restrict__ out, int S)
{
    int idx = blockIdx.x * blockDim.x + threadIdx.x;
    const int total = NB * S * S;
    if (idx >= total) return;
    const int x  = idx % S;
    const int y  = (idx / S) % S;
    const int nb = idx / (S * S);
    const float scale = (float)(HF - 1) / (float)(S - 1);
    float sy = y * scale, sx = x * scale;
    int y0 = (int)floorf(sy), x0 = (int)floorf(sx);
    int y1 = min(y0 + 1, HF - 1), x1 = min(x0 + 1, HF - 1);
    float fy = sy - y0, fx = sx - x0;
    const float* dp = depth + (long long)nb * HF * HF;
    out[idx] = (1.f - fy) * ((1.f - fx) * dp[y0*HF+x0] + fx * dp[y0*HF+x1])
             +        fy  * ((1.f - fx) * dp[y1*HF+x0] + fx * dp[y1*HF+x1]);
}

// --------------------------------------------------------------------------
// Fused triple softmax: S = softmax(softmax(scores) * softmax(Rd_logits))
// per row; Rd_logits computed on the fly from d1[row], d2[col].
// One 256-thread block per row of V=1024 columns (float4 each).
// --------------------------------------------------------------------------
#define SMB 256

__device__ __forceinline__ float block_max(float v, float* red)
{
    red[threadIdx.x] = v; __syncthreads();
    for (int s = SMB / 2; s > 0; s >>= 1) {
        if ((int)threadIdx.x < s)
            red[threadIdx.x] = fmaxf(red[threadIdx.x], red[threadIdx.x + s]);
        __syncthreads();
    }
    float r = red[0]; __syncthreads();
    return r;
}

__device__ __forceinline__ float block_sum(float v, float* red)
{
    red[threadIdx.x] = v; __syncthreads();
    for (int s = SMB / 2; s > 0; s >>= 1) {
        if ((int)threadIdx.x < s)
            red[threadIdx.x] += red[threadIdx.x + s];
        __syncthreads();
    }
    float r = red[0]; __syncthreads();
    return r;
}

__global__ void attn_softmax_kernel(float* __restrict__ scores,
                                    const float* __restrict__ d1,
                                    const float* __restrict__ d2)
{
    __shared__ float red[SMB];
    const int row = blockIdx.x;              // 0 .. NB*QN-1
    const int nb  = row / QN;
    const int t   = threadIdx.x;
    float* srow = scores + (long long)row * VN;

    float4 ld = ((const float4*)srow)[t];
    float s[4] = {ld.x, ld.y, ld.z, ld.w};

    // --- Ra = softmax(scores) ---
    float m1 = block_max(fmaxf(fmaxf(s[0], s[1]), fmaxf(s[2], s[3])), red);
    float ra[4], ls = 0.f;
    #pragma unroll
    for (int j = 0; j < 4; ++j) { ra[j] = expf(s[j] - m1); ls += ra[j]; }
    float sum1 = block_sum(ls, red);
    #pragma unroll
    for (int j = 0; j < 4; ++j) ra[j] /= sum1;

    // --- Rd = softmax(min(d1/(d2+eps), d2/(d1+eps))) ---
    const float dq = d1[row];
    float rd[4];
    #pragma unroll
    for (int j = 0; j < 4; ++j) {
        float dv = d2[nb * VN + t * 4 + j];
        rd[j] = fminf(dq / (dv + EPS_D), dv / (dq + EPS_D));
    }
    float m2 = block_max(fmaxf(fmaxf(rd[0], rd[1]), fmaxf(rd[2], rd[3])), red);
    ls = 0.f;
    #pragma unroll
    for (int j = 0; j < 4; ++j) { rd[j] = expf(rd[j] - m2); ls += rd[j]; }
    float sum2 = block_sum(ls, red);
    #pragma unroll
    for (int j = 0; j < 4; ++j) rd[j] /= sum2;

    // --- S = softmax(Ra * Rd) ---
    float tt[4];
    #pragma unroll
    for (int j = 0; j < 4; ++j) tt[j] = ra[j] * rd[j];
    float m3 = block_max(fmaxf(fmaxf(tt[0], tt[1]), fmaxf(tt[2], tt[3])), red);
    ls = 0.f;
    #pragma unroll
    for (int j = 0; j < 4; ++j) { tt[j] = expf(tt[j] - m3); ls += tt[j]; }
    float sum3 = block_sum(ls, red);

    float4 st;
    st.x = tt[0] / sum3; st.y = tt[1] / sum3;
    st.z = tt[2] / sum3; st.w = tt[3] / sum3;
    ((float4*)srow)[t] = st;
}

// out = x + bilinear_up(z, 64->256, align_corners=True)
__global__ void final_kernel(const float* __restrict__ x,
                             const float* __restrict__ z,
                             float* __restrict__ out)
{
    int idx = blockIdx.x * blockDim.x + threadIdx.x;
    const int total = NB * CIN * HF * HF;
    if (idx >= total) return;
    const int xx = idx % HF;
    const int yy = (idx / HF) % HF;
    const long long nc = idx / (HF * HF);
    const float scale = (float)(HD - 1) / (float)(HF - 1);
    float sy = yy * scale, sx = xx * scale;
    int y0 = (int)floorf(sy), x0 = (int)floorf(sx);
    int y1 = min(y0 + 1, HD - 1), x1 = min(x0 + 1, HD - 1);
    float fy = sy - y0, fx = sx - x0;
    const float* zp = z + nc * QN;
    float v = (1.f - fy) * ((1.f - fx) * zp[y0*HD+x0] + fx * zp[y0*HD+x1])
            +        fy  * ((1.f - fx) * zp[y1*HD+x0] + fx * zp[y1*HD+x1]);
    out[idx] = x[idx] + v;
}

// --------------------------------------------------------------------------
// Host side
// --------------------------------------------------------------------------
static void gemm(hipStream_t s,
                 const float* A, long long ars, long long acs, long long abs_,
                 const float* B, long long brs, long long bcs, long long bbs,
                 const float* bias,
                 float* C, long long crs, long long ccs, long long cbs,
                 int M, int N, int K, int batches)
{
    // 4 waves/block, each wave computes a 16x64 strip (NT=4). N % 256 == 0 here.
    dim3 grid(N / 256, M / 16, batches);
    gemm16_wmma_f32<4><<<grid, 128, 0, s>>>(A, ars, acs, abs_,
                                            B, brs, bcs, bbs,
                                            bias, C, crs, ccs, cbs, M, N, K);
}

extern "C" void kernel_launch(void* const* d_in, const int* in_sizes, int n_in,
                              void* d_out, int out_size, void* d_ws, size_t ws_size,
                              hipStream_t stream)
{
    (void)in_sizes; (void)n_in; (void)out_size; (void)ws_size;
    const float* x     = (const float*)d_in[0];
    const float* depth = (const float*)d_in[1];
    const float* dw    = (const float*)d_in[2];
    const float* th_w  = (const float*)d_in[3];
    const float* th_b  = (const float*)d_in[4];
    const float* ph_w  = (const float*)d_in[5];
    const float* ph_b  = (const float*)d_in[6];
    const float* g_w   = (const float*)d_in[7];
    const float* g_b   = (const float*)d_in[8];
    const float* z_w   = (const float*)d_in[9];
    const float* z_b   = (const float*)d_in[10];
    struct Pyr { const float *pw, *gamma, *beta, *ow, *ob; };
    Pyr ppg = { (const float*)d_in[11], (const float*)d_in[12], (const float*)d_in[13],
                (const float*)d_in[14], (const float*)d_in[15] };
    Pyr ppt = { (const float*)d_in[16], (const float*)d_in[17], (const float*)d_in[18],
                (const float*)d_in[19], (const float*)d_in[20] };
    Pyr ppp = { (const float*)d_in[21], (const float*)d_in[22], (const float*)d_in[23],
                (const float*)d_in[24], (const float*)d_in[25] };

    // workspace layout (floats)
    float* ws = (float*)d_ws;
    size_t off = 0;
    auto alloc = [&](size_t cnt) { float* p = ws + off; off += cnt; return p; };
    float* xd    = alloc((size_t)NB * CIN * QN);     // x downsampled
    float* tpre  = alloc((size_t)NB * CC * QN);      // theta conv (pre pyramid)
    float* ppre  = alloc((size_t)NB * CC * QN);      // phi conv  (pre pool)
    float* gpre  = alloc((size_t)NB * CC * QN);      // g conv    (pre pool)
    float* ppool = alloc((size_t)NB * CC * VN);      // phi maxpooled
    float* gpool = alloc((size_t)NB * CC * VN);      // g maxpooled
    float* pools = alloc((size_t)3 * NB * CC * (64 + 16 + 4 + 1));  // avg pools, 3 branches
    float* pyr   = alloc((size_t)3 * NB * 16 * (64 + 16 + 4 + 1));  // pyramid feats
    float* cat_t = alloc((size_t)NB * 2 * CC * QN);
    float* cat_p = alloc((size_t)NB * 2 * CC * VN);
    float* cat_g = alloc((size_t)NB * 2 * CC * VN);
    float* theta = alloc((size_t)NB * CC * QN);      // [n, cc, Q]
    float* phi   = alloc((size_t)NB * CC * VN);      // [n, cc, V]
    float* gbuf  = alloc((size_t)NB * CC * VN);      // [n, cc, V]
    float* dq    = alloc((size_t)NB * QN);           // d1
    float* dv    = alloc((size_t)NB * VN);           // d2
    float* sc    = alloc((size_t)NB * QN * VN);      // scores / S (64 MB)
    float* ybuf  = alloc((size_t)NB * CC * QN);      // y^T [n, cc, Q]
    float* zbuf  = alloc((size_t)NB * CIN * QN);     // z conv output

    // 1) depthwise 4x4 stride-4 downsample
    down_conv_kernel<<<(NB*CIN*QN + 255) / 256, 256, 0, stream>>>(x, dw, xd);

    // 2) theta/phi/g 1x1 convs: [CC,CIN] x [CIN,QN] per batch (WMMA)
    gemm(stream, th_w, CIN, 1, 0, xd, QN, 1, (long long)CIN*QN, th_b,
         tpre, QN, 1, (long long)CC*QN, CC, QN, CIN, NB);
    gemm(stream, ph_w, CIN, 1, 0, xd, QN, 1, (long long)CIN*QN, ph_b,
         ppre, QN, 1, (long long)CC*QN, CC, QN, CIN, NB);
    gemm(stream, g_w,  CIN, 1, 0, xd, QN, 1, (long long)CIN*QN, g_b,
         gpre, QN, 1, (long long)CC*QN, CC, QN, CIN, NB);

    // 3) maxpool phi and g branches to 32x32
    maxpool2_kernel<<<(NB*CC*VN + 255) / 256, 256, 0, stream>>>(ppre, ppool, HD);
    maxpool2_kernel<<<(NB*CC*VN + 255) / 256, 256, 0, stream>>>(gpre, gpool, HD);

    // 4) pyramid pooling on three branches
    const int psl[4] = {8, 4, 2, 1};
    const int pool_off[4] = {0, NB*CC*64, NB*CC*(64+16), NB*CC*(64+16+4)};
    const int pyr_off[4]  = {0, NB*16*64, NB*16*(64+16), NB*16*(64+16+4)};
    struct Branch { const float* in; int H; const Pyr* pp; float* cat; float* out; int hw; };
    Branch brs[3] = {
        { tpre,  HD, &ppt, cat_t, theta, QN },
        { ppool, HQ, &ppp, cat_p, phi,   VN },
        { gpool, HQ, &ppg, cat_g, gbuf,  VN },
    };
    for (int b = 0; b < 3; ++b) {
        float* poolb = pools + (size_t)b * NB * CC * (64 + 16 + 4 + 1);
        float* pyrb  = pyr   + (size_t)b * NB * 16 * (64 + 16 + 4 + 1);
        for (int i = 0; i < 4; ++i) {
            const int ps = psl[i], psq = ps * ps;
            avgpool_kernel<<<(NB*CC*psq + 255) / 256, 256, 0, stream>>>(
                brs[b].in, poolb + pool_off[i], brs[b].H, ps);
            pyrconv_kernel<<<(NB*16*psq + 255) / 256, 256, 0, stream>>>(
                poolb + pool_off[i], brs[b].pp->pw + i * 16 * CC,
                brs[b].pp->gamma + i * 16, brs[b].pp->beta + i * 16,
                pyrb + pyr_off[i], psq);
        }
        upcat_kernel<<<(NB*2*CC*brs[b].hw + 255) / 256, 256, 0, stream>>>(
            brs[b].in, pyrb, brs[b].cat, brs[b].H);
        // output conv: [CC, 2CC] x [2CC, hw] + ob   (WMMA)
        gemm(stream, brs[b].pp->ow, 2*CC, 1, 0,
             brs[b].cat, brs[b].hw, 1, (long long)2*CC*brs[b].hw,
             brs[b].pp->ob,
             brs[b].out, brs[b].hw, 1, (long long)CC*brs[b].hw,
             CC, brs[b].hw, 2*CC, NB);
    }

    // 5) depth resizes (align_corners=True)
    depth_resize_kernel<<<(NB*QN + 255) / 256, 256, 0, stream>>>(depth, dq, HD);
    depth_resize_kernel<<<(NB*VN + 255) / 256, 256, 0, stream>>>(depth, dv, HQ);

    // 6) scores = theta^T @ phi : A[m=q][k=c] = theta[c*QN+q] (col-major via strides)
    gemm(stream, theta, 1, QN, (long long)CC*QN,
         phi, VN, 1, (long long)CC*VN,
         nullptr, sc, VN, 1, (long long)QN*VN,
         QN, VN, CC, NB);

    // 7) fused Ra -> Rd -> S softmax, in place on score buffer
    attn_softmax_kernel<<<NB*QN, SMB, 0, stream>>>(sc, dq, dv);

    // 8) y^T = g @ S^T : C[c][q] = sum_v g[c][v] * S[q][v]
    //    (M=CC, N=QN keeps NT=4 legal and makes step 9's B reads unit-stride)
    gemm(stream, gbuf, VN, 1, (long long)CC*VN,      // A[m=c][k=v]
         sc, 1, VN, (long long)QN*VN,                // B[k=v][n=q] = S[q*VN+v]
         nullptr, ybuf, QN, 1, (long long)CC*QN,     // C[c][q] -> y^T [n, cc, Q]
         CC, QN, VN, NB);

    // 9) z = z_w @ y^T + z_b : B[k=c][n=p] = ybuf[c*QN+p] (coalesced)
    gemm(stream, z_w, CC, 1, 0,
         ybuf, QN, 1, (long long)CC*QN,
         z_b, zbuf, QN, 1, (long long)CIN*QN,
         CIN, QN, CC, NB);

    // 10) out = x + upsample(z, 64->256, align_corners=True)
    final_kernel<<<(NB*CIN*HF*HF + 255) / 256, 256, 0, stream>>>(
        x, zbuf, (float*)d_out);
}